// TemporalSpikingMechanism_57140244906576
// MI455X (gfx1250) — compile-verified
//
#include <hip/hip_runtime.h>

typedef __attribute__((ext_vector_type(2))) float v2f;
typedef __attribute__((ext_vector_type(8))) float v8f;

#define BSZ    32
#define CCH    128     // output channels (and x input channels)
#define K2     256     // conv_w row length (2*C)
#define HWSZ   1024    // H*W
#define TSTEPS 8

// Kernel 0: fold the time-embedding half of the 1x1 conv into a per-(b,o) bias.
// bias2[b][o] = conv_b[o] + sum_i conv_w[o][128+i] * embed_table[t[b]][i]
__global__ void snn_bias2_kernel(const int* __restrict__ t,
                                 const float* __restrict__ embed,
                                 const float* __restrict__ conv_w,
                                 const float* __restrict__ conv_b,
                                 float* __restrict__ bias2) {
  int idx = blockIdx.x * blockDim.x + threadIdx.x;   // 0 .. 4095
  int b = idx >> 7;
  int o = idx & (CCH - 1);
  const float* e = embed + (long)t[b] * CCH;
  const float* w = conv_w + (long)o * K2 + CCH;      // second half of the row
  float acc = conv_b[o];
#pragma unroll 8
  for (int i = 0; i < CCH; ++i) acc = fmaf(w[i], e[i], acc);
  bias2[idx] = acc;
}

// Kernel 1: fused  W[:, :128] @ x[b]  (fp32 WMMA)  + bias2  + 8-step LIF + spike store.
// One 16x16 output tile per wave32. M = output channel, N = spatial index.
__global__ void __launch_bounds__(256)
snn_gemm_lif_kernel(const float* __restrict__ x,
                    const float* __restrict__ conv_w,
                    const float* __restrict__ bias2,
                    float* __restrict__ out) {
  const int lane  = threadIdx.x & 31;
  const int wave  = threadIdx.x >> 5;
  const int tile  = blockIdx.x * 8 + wave;   // 16384 tiles total
  const int b     = tile >> 9;               // / (8*64)
  const int rem   = tile & 511;
  const int m0    = (rem >> 6) << 4;         // 8 M-tiles of 16
  const int n0    = (rem & 63) << 4;         // 64 N-tiles of 16

  const int lhalf = lane >> 4;               // 0: K{0,1}/lanes hold M,N 0..15 ; 1: K{2,3}
  const int l16   = lane & 15;

  // A-matrix 16x4 fp32: lane l16 -> row M=m0+l16; vgpr pair holds K = 2*lhalf, 2*lhalf+1
  const float* wrow = conv_w + (long)(m0 + l16) * K2;
  // B-matrix 4x16 fp32: lane l16 -> col N=n0+l16; same K split per lane half
  const float* xcol = x + (long)b * CCH * HWSZ + n0 + l16;

  v8f c = {};
#pragma unroll 4
  for (int k = 0; k < CCH; k += 4) {
    const int kk = k + 2 * lhalf;
    v2f a, bm;
    a.x  = wrow[kk];
    a.y  = wrow[kk + 1];
    bm.x = xcol[(long)kk * HWSZ];
    bm.y = xcol[(long)(kk + 1) * HWSZ];
    // 8 args: (neg_a, A, neg_b, B, c_mod, C, reuse_a, reuse_b)
    c = __builtin_amdgcn_wmma_f32_16x16x4_f32(false, a, false, bm,
                                              (short)0, c, false, false);
  }

  // C/D layout: vgpr r -> M = r + 8*lhalf (rows 0..15), N = l16
  const int s = n0 + l16;
  const float* bb = bias2 + b * CCH + m0;
  float* outb = out + (long)b * TSTEPS * CCH * HWSZ;

#pragma unroll
  for (int r = 0; r < 8; ++r) {
    const int   M   = r + 8 * lhalf;
    const float cur = c[r] + bb[M];
    float* po = outb + (long)(m0 + M) * HWSZ + s;
    float v = 0.0f;
#pragma unroll
    for (int tt = 0; tt < TSTEPS; ++tt) {
      v = 0.5f * (v + cur);                       // v += (cur - v)/TAU, TAU=2
      const float spike = (v >= 1.0f) ? 1.0f : 0.0f;  // forward of surrogate = heaviside
      po[(long)tt * CCH * HWSZ] = spike;
      v -= spike;                                  // soft reset, V_TH=1
    }
  }
}

extern "C" void kernel_launch(void* const* d_in, const int* in_sizes, int n_in,
                              void* d_out, int out_size, void* d_ws, size_t ws_size,
                              hipStream_t stream) {
  const float* x      = (const float*)d_in[0];
  const int*   t      = (const int*)  d_in[1];
  const float* embed  = (const float*)d_in[2];
  const float* conv_w = (const float*)d_in[3];
  const float* conv_b = (const float*)d_in[4];
  float*       out    = (float*)d_out;
  float*       bias2  = (float*)d_ws;   // 32*128 floats = 16 KB scratch

  snn_bias2_kernel<<<(BSZ * CCH) / 256, 256, 0, stream>>>(t, embed, conv_w, conv_b, bias2);
  snn_gemm_lif_kernel<<<(BSZ * 8 * 64) / 8, 256, 0, stream>>>(x, conv_w, bias2, out);
}